// EMAQuantizer_6399501271152
// MI455X (gfx1250) — compile-verified
//
#include <hip/hip_runtime.h>

typedef __attribute__((ext_vector_type(2))) float v2f;
typedef __attribute__((ext_vector_type(8))) float v8f;

#define EMBED_D   256
#define NCODES    8192
#define NROWS     16384
#define ROWS_WG   64
#define THREADS_WG 128          // 4 waves (wave32)
#define KCHUNK    16
#define LSTRIDE   260           // 256 + 4 pad: 16B-aligned rows, conflict-free wmma frags

// ---------------------------------------------------------------------------
// Kernel 1: row squared-norms (one wave32 per 256-wide row)
// ---------------------------------------------------------------------------
__global__ void rownorm_kernel(const float* __restrict__ x, float* __restrict__ out, int nrows) {
  int wave = threadIdx.x >> 5, lane = threadIdx.x & 31;
  int row = blockIdx.x * 8 + wave;
  if (row >= nrows) return;
  const float4* p = (const float4*)(x + (size_t)row * EMBED_D);
  float s = 0.f;
#pragma unroll
  for (int c = 0; c < 2; ++c) {
    float4 v = p[lane + 32 * c];
    s += v.x * v.x + v.y * v.y + v.z * v.z + v.w * v.w;
  }
#pragma unroll
  for (int o = 16; o > 0; o >>= 1) s += __shfl_xor(s, o, 32);
  if (lane == 0) out[row] = s;
}

// ---------------------------------------------------------------------------
// Kernel 2: fused distance-GEMM (V_WMMA_F32_16X16X4_F32) + argmin + gather +
//           loss partial sums + counts/dw scatter.
// A = weight chunk (16 codes x D), B = z tile (D x 16 z-rows). C[m][n] puts
// all 16 code-distances of z-row n into lane n (+ lane n+16), so argmin is a
// per-lane running scalar min.
// ---------------------------------------------------------------------------
__global__ void __launch_bounds__(THREADS_WG)
vq_main_kernel(const float* __restrict__ z, const float* __restrict__ weight,
               const float* __restrict__ wnorm, const float* __restrict__ znorm,
               float* __restrict__ counts, float* __restrict__ dw,
               float* __restrict__ lossSum,
               float* __restrict__ zq_out, int* __restrict__ idx_out)
{
  extern __shared__ float lds[];
  float* sZ = lds;                          // [ROWS_WG][LSTRIDE]
  float* sW = lds + ROWS_WG * LSTRIDE;      // [KCHUNK][LSTRIDE]

  const int tid  = threadIdx.x;
  const int lane = tid & 31;
  const int wave = tid >> 5;
  const int lr   = lane & 15;
  const int half = lane >> 4;
  const int wgRow0   = blockIdx.x * ROWS_WG;
  const int waveRow0 = wgRow0 + wave * 16;

  // stage z tile (64 rows x 256) into LDS as float4s
  for (int i = tid; i < ROWS_WG * (EMBED_D / 4); i += THREADS_WG) {
    int r = i >> 6, c = i & 63;
    float4 v = ((const float4*)(z + (size_t)(wgRow0 + r) * EMBED_D))[c];
    *(float4*)&sZ[r * LSTRIDE + c * 4] = v;
  }

  const float zn = znorm[waveRow0 + lr];
  float bestd = 3.4e38f;
  int   besti = 0;

  const float* aBase = &sW[lr * LSTRIDE + 2 * half];                 // A[m=lr][k = d + 2*half + {0,1}]
  const float* bBase = &sZ[(wave * 16 + lr) * LSTRIDE + 2 * half];   // B[k][n=lr] = z[n][k]

  for (int k0 = 0; k0 < NCODES; k0 += KCHUNK) {
    __syncthreads();  // also covers initial z-tile visibility
    for (int i = tid; i < KCHUNK * (EMBED_D / 4); i += THREADS_WG) {
      int r = i >> 6, c = i & 63;
      float4 v = ((const float4*)(weight + (size_t)(k0 + r) * EMBED_D))[c];
      *(float4*)&sW[r * LSTRIDE + c * 4] = v;
    }
    __syncthreads();

    v8f acc0 = {}, acc1 = {}, acc2 = {}, acc3 = {};   // 4 chains to hide wmma latency
#pragma unroll 4
    for (int d = 0; d < EMBED_D; d += 16) {
      v2f a0 = *(const v2f*)(aBase + d +  0), b0 = *(const v2f*)(bBase + d +  0);
      v2f a1 = *(const v2f*)(aBase + d +  4), b1 = *(const v2f*)(bBase + d +  4);
      v2f a2 = *(const v2f*)(aBase + d +  8), b2 = *(const v2f*)(bBase + d +  8);
      v2f a3 = *(const v2f*)(aBase + d + 12), b3 = *(const v2f*)(bBase + d + 12);
      acc0 = __builtin_amdgcn_wmma_f32_16x16x4_f32(false, a0, false, b0, (short)0, acc0, false, false);
      acc1 = __builtin_amdgcn_wmma_f32_16x16x4_f32(false, a1, false, b1, (short)0, acc1, false, false);
      acc2 = __builtin_amdgcn_wmma_f32_16x16x4_f32(false, a2, false, b2, (short)0, acc2, false, false);
      acc3 = __builtin_amdgcn_wmma_f32_16x16x4_f32(false, a3, false, b3, (short)0, acc3, false, false);
    }

#pragma unroll
    for (int i = 0; i < 8; ++i) {
      int m = half * 8 + i;  // C layout: lanes 0-15 hold M=i, lanes 16-31 hold M=8+i
      float dot = (acc0[i] + acc1[i]) + (acc2[i] + acc3[i]);
      float dist = zn + wnorm[k0 + m] - 2.0f * dot;
      if (dist < bestd) { bestd = dist; besti = k0 + m; }
    }
  }

  // combine the two lane-halves (codes m<8 vs m>=8 per chunk); tie -> lower idx
  {
    float od = __shfl_xor(bestd, 16, 32);
    int   oi = __shfl_xor(besti, 16, 32);
    if (od < bestd || (od == bestd && oi < besti)) { bestd = od; besti = oi; }
  }

  if (lane < 16) {
    idx_out[waveRow0 + lr] = besti;
    unsafeAtomicAdd(&counts[besti], 1.0f);
  }

  // gather z_q, straight-through output z + (w - z), loss, dw scatter
  float lossAcc = 0.f;
  for (int r = 0; r < 16; ++r) {
    int bi = __shfl(besti, r, 32);
    const float4* wrow = (const float4*)(weight + (size_t)bi * EMBED_D);
    float4* zq4 = (float4*)(zq_out + (size_t)(waveRow0 + r) * EMBED_D);
    const float* zrow = &sZ[(wave * 16 + r) * LSTRIDE];
    float* dwrow = dw + (size_t)bi * EMBED_D;
#pragma unroll
    for (int cc = 0; cc < 2; ++cc) {
      int c = lane + 32 * cc;
      float4 w4 = wrow[c];
      float4 z4 = *(const float4*)&zrow[c * 4];
      float4 t, o;
      t.x = w4.x - z4.x; t.y = w4.y - z4.y; t.z = w4.z - z4.z; t.w = w4.w - z4.w;
      o.x = z4.x + t.x;  o.y = z4.y + t.y;  o.z = z4.z + t.z;  o.w = z4.w + t.w;
      zq4[c] = o;
      lossAcc += t.x * t.x + t.y * t.y + t.z * t.z + t.w * t.w;
      unsafeAtomicAdd(&dwrow[c * 4 + 0], z4.x);
      unsafeAtomicAdd(&dwrow[c * 4 + 1], z4.y);
      unsafeAtomicAdd(&dwrow[c * 4 + 2], z4.z);
      unsafeAtomicAdd(&dwrow[c * 4 + 3], z4.w);
    }
  }
#pragma unroll
  for (int o = 16; o > 0; o >>= 1) lossAcc += __shfl_xor(lossAcc, o, 32);
  if (lane == 0) unsafeAtomicAdd(lossSum, lossAcc);
}

// ---------------------------------------------------------------------------
// Kernel 3a: new_cluster_size, its total n, and loss finalize (1 block)
// ---------------------------------------------------------------------------
__global__ void finalize_cs_kernel(const float* __restrict__ cluster_size,
                                   const float* __restrict__ counts,
                                   float* __restrict__ out_cs,
                                   float* __restrict__ nOut,
                                   const float* __restrict__ lossSum,
                                   float* __restrict__ lossOut)
{
  __shared__ float red[32];
  int tid = threadIdx.x;                 // 1024 threads = 32 waves
  float s = 0.f;
  for (int k = tid; k < NCODES; k += 1024) {
    float v = 0.99f * cluster_size[k] + 0.01f * counts[k];
    out_cs[k] = v;
    s += v;
  }
#pragma unroll
  for (int o = 16; o > 0; o >>= 1) s += __shfl_xor(s, o, 32);
  int lane = tid & 31, wave = tid >> 5;
  if (lane == 0) red[wave] = s;
  __syncthreads();
  if (wave == 0) {
    float v = red[lane];
#pragma unroll
    for (int o = 16; o > 0; o >>= 1) v += __shfl_xor(v, o, 32);
    if (lane == 0) {
      nOut[0] = v;
      lossOut[0] = lossSum[0] * (1.0f / ((float)NROWS * (float)EMBED_D));
    }
  }
}

// ---------------------------------------------------------------------------
// Kernel 3b: new_embed_avg and normalized new_weight
// ---------------------------------------------------------------------------
__global__ void finalize_w_kernel(const float* __restrict__ embed_avg,
                                  const float* __restrict__ dw,
                                  const float* __restrict__ new_cs,
                                  const float* __restrict__ nPtr,
                                  float* __restrict__ out_w,
                                  float* __restrict__ out_ea)
{
  int k = blockIdx.x;
  int d = threadIdx.x;
  size_t i = (size_t)k * EMBED_D + d;
  float ea = 0.99f * embed_avg[i] + 0.01f * dw[i];
  out_ea[i] = ea;
  float n = nPtr[0];
  float cs = (new_cs[k] + 1e-5f) / (n + (float)NCODES * 1e-5f) * n;
  out_w[i] = ea / cs;
}

// ---------------------------------------------------------------------------
extern "C" void kernel_launch(void* const* d_in, const int* in_sizes, int n_in,
                              void* d_out, int out_size, void* d_ws, size_t ws_size,
                              hipStream_t stream)
{
  const float* z            = (const float*)d_in[0];
  const float* weight       = (const float*)d_in[1];
  const float* cluster_size = (const float*)d_in[2];
  const float* embed_avg    = (const float*)d_in[3];

  // output layout (concatenated flat, reference return order)
  float* out      = (float*)d_out;
  float* out_zq   = out;                         // 16*32*32*256 = 4194304
  float* out_loss = out + 4194304;               // 1
  int*   out_idx  = (int*)(out + 4194305);       // 16384 (int32)
  float* out_w    = out + 4210689;               // 2097152
  float* out_cs   = out + 6307841;               // 8192
  float* out_ea   = out + 6316033;               // 2097152

  // workspace layout (floats)
  float* ws      = (float*)d_ws;
  float* wnorm   = ws;                 // 8192
  float* znorm   = ws + 8192;          // 16384
  float* counts  = ws + 24576;         // 8192   (zeroed)
  float* dw      = ws + 32768;         // 2097152 (zeroed)
  float* lossSum = ws + 2129920;       // 1      (zeroed)
  float* nSum    = ws + 2129921;       // 1      (zeroed)

  hipMemsetAsync(counts, 0, (size_t)(8192 + 2097152 + 2) * sizeof(float), stream);

  rownorm_kernel<<<NCODES / 8, 256, 0, stream>>>(weight, wnorm, NCODES);
  rownorm_kernel<<<NROWS / 8, 256, 0, stream>>>(z, znorm, NROWS);

  const size_t ldsBytes = (size_t)(ROWS_WG * LSTRIDE + KCHUNK * LSTRIDE) * sizeof(float); // 83200
  (void)hipFuncSetAttribute((const void*)vq_main_kernel,
                            hipFuncAttributeMaxDynamicSharedMemorySize, (int)ldsBytes);
  vq_main_kernel<<<NROWS / ROWS_WG, THREADS_WG, ldsBytes, stream>>>(
      z, weight, wnorm, znorm, counts, dw, lossSum, out_zq, out_idx);

  finalize_cs_kernel<<<1, 1024, 0, stream>>>(cluster_size, counts, out_cs, nSum, lossSum, out_loss);
  finalize_w_kernel<<<NCODES, EMBED_D, 0, stream>>>(embed_avg, dw, out_cs, nSum, out_w, out_ea);
}